// DynModel_7911329759724
// MI455X (gfx1250) — compile-verified
//
#include <hip/hip_runtime.h>
#include <hip/hip_bf16.h>
#include <math.h>

// ---------------------------------------------------------------------------
// Unicycle rollout = 3 prefix sums + elementwise trig. Memory-bound (~44MB
// total HBM traffic -> ~1.9us at 23.3 TB/s); launch overhead dominates, so
// the pipeline is squeezed to 5 kernels:
//   K0: copy controls->out fused with per-block sums of dth   (1 read, 1 write)
//   K1: single-block exclusive scan of dth block sums (+th0, writes traj[0])
//   K2: WMMA theta scan, writes theta, emits block sums of dx/dy
//   K3: 2-block scan of dx/dy block sums (+x0/y0)
//   K4: recompute theta_prev via WMMA, compute dx/dy, 2 WMMA scans, write x,y
// Intra-wave scans run on the matrix unit: exclusive prefix = (strict
// lower-triangular ones) x (data) via chained V_WMMA_F32_16X16X4_F32
// (K=16 in 4 slices), f32 in / f32 acc. One wave scans a 16x16 tile
// (256 elems, one 16-seg per column). Cross-column: 16-lane shuffle scan.
// Cross-wave: LDS. Cross-block: block-partial scan kernels.
// ---------------------------------------------------------------------------

typedef __attribute__((ext_vector_type(2))) float v2f;
typedef __attribute__((ext_vector_type(8))) float v8f;

#define WAVE 32
#define CHUNK 2048            // elements per 256-thread block (8 waves x 256)

__device__ __forceinline__ v8f wmma_k4(v2f a, v2f b, v8f c) {
  // D = A(16x4) x B(4x16) + C(16x16), all f32
  return __builtin_amdgcn_wmma_f32_16x16x4_f32(
      /*neg_a=*/false, a, /*neg_b=*/false, b,
      /*c_mod=*/(short)0, c, /*reuse_a=*/false, /*reuse_b=*/false);
}

// A-operand slice t of strict-lower-triangular ones L'[m][k] = (k < m).
// ISA layout (16x4 f32 A): lanes 0-15: M=lane, VGPR0=K0,VGPR1=K1;
//                          lanes 16-31: M=lane-16, VGPR0=K2,VGPR1=K3.
__device__ __forceinline__ v2f tri_sliceA(int t, int lane) {
  int row = lane & 15;
  int k0 = 4 * t + ((lane >= 16) ? 2 : 0);
  v2f a;
  a.x = (k0 < row) ? 1.0f : 0.0f;
  a.y = (k0 + 1 < row) ? 1.0f : 0.0f;
  return a;
}

// Exclusive in-column scan of a 16x16 tile X (column j = one 16-elem segment):
// D[m][j] = sum_{k<m} X[k][j].  b[t] holds K-slice 4t..4t+3 in B layout
// (lane n<16: rows 4t,4t+1 of column n; lane n>=16: rows 4t+2,4t+3).
__device__ __forceinline__ v8f wmma_excl_scan(const v2f b[4], int lane) {
  v8f acc = {};
  acc = wmma_k4(tri_sliceA(0, lane), b[0], acc);
  acc = wmma_k4(tri_sliceA(1, lane), b[1], acc);
  acc = wmma_k4(tri_sliceA(2, lane), b[2], acc);
  acc = wmma_k4(tri_sliceA(3, lane), b[3], acc);
  return acc;
}

// Cross-column exclusive offsets. colTotalHi holds column totals in lanes
// 16..31 (column = lane-16). Returns exclusive prefix for this lane's column
// (column = lane&15); *waveTotal = total over all 16 columns.
__device__ __forceinline__ float cross_col_excl(float colTotalHi, int lane,
                                                float* waveTotal) {
  float c = __shfl(colTotalHi, 16 + (lane & 15), WAVE);
  float incl = c;
#pragma unroll
  for (int d = 1; d < 16; d <<= 1) {
    float t = __shfl_up(incl, d, 16);
    if ((lane & 15) >= d) incl += t;
  }
  float excl = __shfl_up(incl, 1, 16);
  if ((lane & 15) == 0) excl = 0.0f;
  *waveTotal = __shfl(incl, 15, WAVE);
  return excl;
}

// -------- K0: copy controls -> out fused with per-block sums of dth --------
// Block b copies floats [b*4096, (b+1)*4096) and sums 10*controls[:,1] over
// its CHUNK elements (odd float indices).
__global__ void k_copy_sums(const float* __restrict__ ctrl,
                            float* __restrict__ out,
                            float* __restrict__ sums, int T) {
  const int tid = threadIdx.x;
  const int n2 = 2 * T;                       // total floats
  const int baseq = blockIdx.x * 1024;        // float4 index base
  float s = 0.f;
#pragma unroll
  for (int t = 0; t < 4; ++t) {
    int q = baseq + t * 256 + tid;
    int f = 4 * q;                            // float index
    if (f + 3 < n2) {
      float4 v = *(const float4*)(ctrl + f);
      *(float4*)(out + f) = v;
      s += 10.0f * v.y + 10.0f * v.w;         // c1[e] + c1[e+1]
    } else {
#pragma unroll
      for (int k = 0; k < 4; ++k) {
        if (f + k < n2) {
          float v = ctrl[f + k];
          out[f + k] = v;
          if ((f + k) & 1) s += 10.0f * v;
        }
      }
    }
  }
#pragma unroll
  for (int d = 16; d >= 1; d >>= 1) s += __shfl_xor(s, d, WAVE);
  __shared__ float wsum[8];
  int lane = tid & 31, wave = tid >> 5;
  if (lane == 0) wsum[wave] = s;
  __syncthreads();
  if (tid == 0) {
    float t = 0.f;
    for (int w = 0; w < 8; ++w) t += wsum[w];
    sums[blockIdx.x] = t;
  }
}

// ---- K1/K3: single-block exclusive scan of block sums (+seed) -------------
// Launched with G blocks; block g scans sums[g*n .. g*n+n) into offs[...]
// seeded with pose[comp0+g]. writePose: block 0 thread writes traj[0].
__global__ void k_scan_offsets(const float* __restrict__ sums,
                               float* __restrict__ offs, int n,
                               const float* __restrict__ pose, int comp0,
                               float* __restrict__ traj0, int writePose) {
  __shared__ float tmp[512];
  __shared__ float carry;
  const int tid = threadIdx.x;
  const float* s = sums + (size_t)blockIdx.x * n;
  float* o = offs + (size_t)blockIdx.x * n;
  if (tid == 0) carry = pose[comp0 + blockIdx.x];
  if (writePose && blockIdx.x == 0 && tid < 3) traj0[tid] = pose[tid];
  __syncthreads();
  for (int base = 0; base < n; base += 512) {
    float v = (base + tid < n) ? s[base + tid] : 0.f;
    tmp[tid] = v;
    __syncthreads();
    for (int d = 1; d < 512; d <<= 1) {
      float t = (tid >= d) ? tmp[tid - d] : 0.f;
      __syncthreads();
      tmp[tid] += t;
      __syncthreads();
    }
    float excl = (tid == 0) ? 0.f : tmp[tid - 1];
    if (base + tid < n) o[base + tid] = carry + excl;
    __syncthreads();
    if (tid == 0) carry += tmp[511];
    __syncthreads();
  }
}

// Shared loader: B-layout dth slices + D-layout (c0, dth) per element.
__device__ __forceinline__ void load_tile(const float* __restrict__ ctrl,
                                          int base, int col, bool hi, int T,
                                          v2f b[4], float c0d[8],
                                          float dthd[8]) {
#pragma unroll
  for (int t = 0; t < 4; ++t) {
    int e = base + 16 * col + 4 * t + (hi ? 2 : 0);
    float x0 = 0.f, x1 = 0.f;
    if (e + 1 < T) {
      float4 f = *(const float4*)(ctrl + 2 * e);  // c0[e],c1[e],c0[e+1],c1[e+1]
      x0 = 10.0f * f.y;
      x1 = 10.0f * f.w;
    } else {
      if (e < T) x0 = 10.0f * ctrl[2 * e + 1];
      if (e + 1 < T) x1 = 10.0f * ctrl[2 * e + 3];
    }
    b[t].x = x0;
    b[t].y = x1;
  }
#pragma unroll
  for (int r = 0; r < 8; ++r) {
    int e = base + 16 * col + r + (hi ? 8 : 0);
    float sc = 0.f, dt = 0.f;
    if (e < T) {
      float2 f = *(const float2*)(ctrl + 2 * e);
      sc = f.x;
      dt = 10.0f * f.y;
    }
    c0d[r] = sc;
    dthd[r] = dt;
  }
}

// ---- K2: WMMA theta scan; write theta; emit block sums of dx, dy ----------
__global__ void k_theta(const float* __restrict__ ctrl, float* __restrict__ traj,
                        const float* __restrict__ offTh,
                        float* __restrict__ sumsX, float* __restrict__ sumsY,
                        int T) {
  const int tid = threadIdx.x;
  const int lane = tid & 31;
  const int wave = tid >> 5;
  const int base = blockIdx.x * CHUNK + wave * 256;
  const int col = lane & 15;
  const bool hi = lane >= 16;

  v2f b[4];
  float c0d[8], dthd[8];
  load_tile(ctrl, base, col, hi, T, b, c0d, dthd);

  v8f s = wmma_excl_scan(b, lane);  // exclusive theta increments per column

  // column totals live in hi lanes: excl[m=15] + elem[m=15] (b[3].y in hi)
  float colTot = s[7] + b[3].y;
  float waveTotal;
  float colOff = cross_col_excl(colTot, lane, &waveTotal);

  __shared__ float wth[8], wx[8], wy[8];
  if (lane == 0) wth[wave] = waveTotal;
  __syncthreads();
  float waveOff = 0.f;
  for (int w = 0; w < wave; ++w) waveOff += wth[w];

  const float off = offTh[blockIdx.x] + waveOff + colOff;

  float sx = 0.f, sy = 0.f;
#pragma unroll
  for (int r = 0; r < 8; ++r) {
    int e = base + 16 * col + r + (hi ? 8 : 0);
    float th_prev = off + s[r];
    float th = th_prev + dthd[r];
    float sp = fabsf(c0d[r]);            // 0 out of range -> dx=dy=0
    sx += sp * cosf(th_prev);
    sy += sp * sinf(th_prev);
    if (e < T) traj[3 * (e + 1) + 2] = th;
  }
#pragma unroll
  for (int d = 16; d >= 1; d >>= 1) {
    sx += __shfl_xor(sx, d, WAVE);
    sy += __shfl_xor(sy, d, WAVE);
  }
  if (lane == 0) { wx[wave] = sx; wy[wave] = sy; }
  __syncthreads();
  if (tid == 0) {
    float tx = 0.f, ty = 0.f;
    for (int w = 0; w < 8; ++w) { tx += wx[w]; ty += wy[w]; }
    sumsX[blockIdx.x] = tx;
    sumsY[blockIdx.x] = ty;
  }
}

// ---- K4: recompute theta_prev (WMMA), dx/dy, WMMA scans for x,y -----------
__global__ void k_xy(const float* __restrict__ ctrl, float* __restrict__ traj,
                     const float* __restrict__ offTh,
                     const float* __restrict__ offX,
                     const float* __restrict__ offY, int T) {
  const int tid = threadIdx.x;
  const int lane = tid & 31;
  const int wave = tid >> 5;
  const int base = blockIdx.x * CHUNK + wave * 256;
  const int col = lane & 15;
  const bool hi = lane >= 16;

  v2f b[4];
  float c0d[8], dthd[8];
  load_tile(ctrl, base, col, hi, T, b, c0d, dthd);

  v8f s = wmma_excl_scan(b, lane);
  float colTot = s[7] + b[3].y;
  float waveTotal;
  float colOff = cross_col_excl(colTot, lane, &waveTotal);

  __shared__ float wth[8], wx[8], wy[8];
  if (lane == 0) wth[wave] = waveTotal;
  __syncthreads();
  float waveOffT = 0.f;
  for (int w = 0; w < wave; ++w) waveOffT += wth[w];
  const float offt = offTh[blockIdx.x] + waveOffT + colOff;

  // dx, dy in D layout: lane holds M = r (+8 if hi), column = lane&15
  float g[8], h[8];
#pragma unroll
  for (int r = 0; r < 8; ++r) {
    float th_prev = offt + s[r];
    float sp = fabsf(c0d[r]);
    g[r] = sp * cosf(th_prev);
    h[r] = sp * sinf(th_prev);
  }

  // D-layout -> B-layout transpose: columns stay in lane pair (L, L^16).
  float pg[8], ph[8];
#pragma unroll
  for (int r = 0; r < 8; ++r) {
    pg[r] = __shfl_xor(g[r], 16, WAVE);
    ph[r] = __shfl_xor(h[r], 16, WAVE);
  }
  v2f bx[4], by[4];
  bx[0].x = hi ? pg[2] : g[0];  bx[0].y = hi ? pg[3] : g[1];
  bx[1].x = hi ? pg[6] : g[4];  bx[1].y = hi ? pg[7] : g[5];
  bx[2].x = hi ? g[2] : pg[0];  bx[2].y = hi ? g[3] : pg[1];
  bx[3].x = hi ? g[6] : pg[4];  bx[3].y = hi ? g[7] : pg[5];
  by[0].x = hi ? ph[2] : h[0];  by[0].y = hi ? ph[3] : h[1];
  by[1].x = hi ? ph[6] : h[4];  by[1].y = hi ? ph[7] : h[5];
  by[2].x = hi ? h[2] : ph[0];  by[2].y = hi ? h[3] : ph[1];
  by[3].x = hi ? h[6] : ph[4];  by[3].y = hi ? h[7] : ph[5];

  v8f sx8 = wmma_excl_scan(bx, lane);
  v8f sy8 = wmma_excl_scan(by, lane);

  float cx = sx8[7] + g[7];  // hi lanes: excl[m=15] + dx[m=15]
  float cy = sy8[7] + h[7];
  float wtx, wty;
  float cOffX = cross_col_excl(cx, lane, &wtx);
  float cOffY = cross_col_excl(cy, lane, &wty);

  if (lane == 0) { wx[wave] = wtx; wy[wave] = wty; }
  __syncthreads();
  float wOffX = 0.f, wOffY = 0.f;
  for (int w = 0; w < wave; ++w) { wOffX += wx[w]; wOffY += wy[w]; }
  const float ox = offX[blockIdx.x] + wOffX + cOffX;
  const float oy = offY[blockIdx.x] + wOffY + cOffY;

#pragma unroll
  for (int r = 0; r < 8; ++r) {
    int e = base + 16 * col + r + (hi ? 8 : 0);
    if (e < T) {
      traj[3 * (e + 1) + 0] = ox + sx8[r] + g[r];
      traj[3 * (e + 1) + 1] = oy + sy8[r] + h[r];
    }
  }
}

extern "C" void kernel_launch(void* const* d_in, const int* in_sizes, int n_in,
                              void* d_out, int out_size, void* d_ws,
                              size_t ws_size, hipStream_t stream) {
  const float* start_pose = (const float*)d_in[0];  // (3,)
  const float* controls = (const float*)d_in[1];    // (T,2) row-major
  const int T = in_sizes[1] / 2;
  float* out = (float*)d_out;
  float* traj = out + (size_t)2 * T;                // (T+1,3)
  const int NB = (T + CHUNK - 1) / CHUNK;

  // ws layout: sums_th | off_th | sums_x | sums_y | off_x | off_y
  // (sums_x,sums_y adjacent so one 2-block scan launch covers both)
  float* w = (float*)d_ws;                          // 6*NB floats (~12KB)
  float* sums_th = w;
  float* off_th = w + NB;
  float* sums_x = w + 2 * NB;
  float* sums_y = w + 3 * NB;
  float* off_x = w + 4 * NB;
  float* off_y = w + 5 * NB;

  // K0: controls passthrough + dth block sums (single read of controls)
  k_copy_sums<<<NB, 256, 0, stream>>>(controls, out, sums_th, T);
  // K1: scan dth block sums, seed th0, write traj[0]
  k_scan_offsets<<<1, 512, 0, stream>>>(sums_th, off_th, NB, start_pose, 2,
                                        traj, 1);
  // K2: theta via WMMA scan; dx/dy block sums
  k_theta<<<NB, 256, 0, stream>>>(controls, traj, off_th, sums_x, sums_y, T);
  // K3: scan x and y block sums in parallel (block 0 -> x, block 1 -> y)
  k_scan_offsets<<<2, 512, 0, stream>>>(sums_x, off_x, NB, start_pose, 0,
                                        nullptr, 0);
  // K4: x,y via WMMA scans
  k_xy<<<NB, 256, 0, stream>>>(controls, traj, off_th, off_x, off_y, T);
}